// MultiHeadAttention_32220844654809
// MI455X (gfx1250) — compile-verified
//
#include <hip/hip_runtime.h>
#include <hip/hip_bf16.h>

// MI455X / gfx1250 implementation.
//
// Algebraic reduction of the reference (see analysis):
//   out = reshape16x( (x @ Wv_sum^T viewed [N,256,2048]) @ Wo^T + bo )
// where Wv_sum[d,e] = sum_h Wv[h*128+d, e]  (head-folded Wv, [128,2048])
// and each of the 2048 unique output rows is broadcast 16x along the
// sequence axis (softmax row-sums == 1 to ~1e-6).
//
// All GEMMs use V_WMMA_F32_16X16X4_F32 (full fp32 precision).

typedef float v2f __attribute__((ext_vector_type(2)));
typedef float v8f __attribute__((ext_vector_type(8)));

#define EMB 2048
#define HEADS 16
#define HD 128            // head dim
#define NB 8              // batch
#define SEQ 4096
#define MROWS (NB * SEQ)  // 32768 rows of x
#define UROWS 2048        // unique output rows (8 * 256)

// ---------------------------------------------------------------------------
// Kernel A: fold Wv over heads.  Wv_sum[d, e] = sum_h Wv[h*128 + d, e]
// 128*2048 = 262144 outputs.
// ---------------------------------------------------------------------------
__global__ __launch_bounds__(256) void fold_wv_kernel(const float* __restrict__ Wv,
                                                      float* __restrict__ Wv_sum) {
    int idx = blockIdx.x * 256 + threadIdx.x;      // 0 .. 128*2048-1
    int d = idx >> 11;                             // 0..127
    int e = idx & (EMB - 1);                       // 0..2047
    float s = 0.0f;
#pragma unroll
    for (int h = 0; h < HEADS; ++h)
        s += Wv[(size_t)(h * HD + d) * EMB + e];
    Wv_sum[idx] = s;                               // row-major [128][2048]
}

// ---------------------------------------------------------------------------
// Kernel B: vsum[m, d] = sum_e x[m, e] * Wv_sum[d, e]
// M = 32768, N = 128, K = 2048.  NT layout, both K-contiguous.
// One wave computes a 16(M) x 128(N) strip: 8 C tiles, shared A fragment.
// fp32 WMMA fragment layout (16x16x4):
//   A (16x4): lane l<16 -> row l, k = {k0,k0+1}; lane l>=16 -> row l-16,
//             k = {k0+2,k0+3}  => one float2 load per lane.
//   B (4x16): symmetric: lane gives column n, same k split.
//   C/D: 8 VGPRs, N = lane%16, M = vgpr + 8*(lane/16).
// ---------------------------------------------------------------------------
__global__ __launch_bounds__(256) void gemm1_wmma_kernel(const float* __restrict__ x,
                                                         const float* __restrict__ Wvs,
                                                         float* __restrict__ vsum) {
    const int wave = (blockIdx.x * 256 + (int)threadIdx.x) >> 5;  // 0..2047
    const int lane = threadIdx.x & 31;
    const int r    = lane & 15;           // row/col within tile
    const int hi   = lane >> 4;           // 0 or 1
    const int kh   = hi << 1;             // k sub-offset: 0 or 2

    const int m0 = wave * 16;             // 16 rows of x per wave
    const float* __restrict__ arow = x + (size_t)(m0 + r) * EMB + kh;

    // 8 B row pointers: n = t*16 + r
    const float* __restrict__ brow[8];
#pragma unroll
    for (int t = 0; t < 8; ++t)
        brow[t] = Wvs + (size_t)(t * 16 + r) * EMB + kh;

    v8f c[8] = {};

    for (int k0 = 0; k0 < EMB; k0 += 4) {
        v2f a = *(const v2f*)(arow + k0);
#pragma unroll
        for (int t = 0; t < 8; ++t) {
            v2f b = *(const v2f*)(brow[t] + k0);
            c[t] = __builtin_amdgcn_wmma_f32_16x16x4_f32(
                false, a, false, b, (short)0, c[t], false, false);
        }
    }

    // store: vsum row-major [32768][128]
#pragma unroll
    for (int t = 0; t < 8; ++t) {
#pragma unroll
        for (int g = 0; g < 8; ++g) {
            int row = m0 + g + (hi << 3);
            vsum[(size_t)row * HD + t * 16 + r] = c[t][g];
        }
    }
}

// ---------------------------------------------------------------------------
// Kernel C: Z[nr, j] = sum_e P[nr, e] * Wo[j, e] + bo[j]
// P = vsum reinterpreted [2048][2048].  M = 2048, N = 2048, K = 2048.
// One wave computes a 16(M) x 64(N) strip (4 tiles), then broadcast-stores
// each Z row 16x:  out[n, qh*256 + rr, j] = Z[n*256 + rr, j], qh = 0..15.
// ---------------------------------------------------------------------------
__global__ __launch_bounds__(256) void gemm2_wmma_kernel(const float* __restrict__ P,
                                                         const float* __restrict__ Wo,
                                                         const float* __restrict__ bo,
                                                         float* __restrict__ out) {
    const int wave = (blockIdx.x * 256 + (int)threadIdx.x) >> 5;  // 0..4095
    const int lane = threadIdx.x & 31;
    const int r    = lane & 15;
    const int hi   = lane >> 4;
    const int kh   = hi << 1;

    const int mt = wave >> 5;         // 0..127  (M tile)
    const int nt = wave & 31;         // 0..31   (N strip of 64)
    const int m0 = mt * 16;
    const int j0 = nt * 64;

    const float* __restrict__ arow = P + (size_t)(m0 + r) * EMB + kh;

    const float* __restrict__ brow[4];
    v8f c[4];
#pragma unroll
    for (int t = 0; t < 4; ++t) {
        int j = j0 + t * 16 + r;
        brow[t] = Wo + (size_t)j * EMB + kh;
        float bias = bo[j];           // bias depends only on column (N)
#pragma unroll
        for (int g = 0; g < 8; ++g) c[t][g] = bias;
    }

    for (int k0 = 0; k0 < EMB; k0 += 4) {
        v2f a = *(const v2f*)(arow + k0);
#pragma unroll
        for (int t = 0; t < 4; ++t) {
            v2f b = *(const v2f*)(brow[t] + k0);
            c[t] = __builtin_amdgcn_wmma_f32_16x16x4_f32(
                false, a, false, b, (short)0, c[t], false, false);
        }
    }

    // broadcast store: each Z row feeds 16 output rows
#pragma unroll
    for (int t = 0; t < 4; ++t) {
        int j = j0 + t * 16 + r;
#pragma unroll
        for (int g = 0; g < 8; ++g) {
            int zrow = m0 + g + (hi << 3);          // 0..2047
            int n  = zrow >> 8;                     // batch
            int rr = zrow & 255;                    // row within period
            float val = c[t][g];
            float* base = out + (size_t)n * SEQ * EMB + (size_t)rr * EMB + j;
            for (int qh = 0; qh < HEADS; ++qh)
                base[(size_t)qh * 256 * EMB] = val;
        }
    }
}

extern "C" void kernel_launch(void* const* d_in, const int* in_sizes, int n_in,
                              void* d_out, int out_size, void* d_ws, size_t ws_size,
                              hipStream_t stream) {
    const float* x   = (const float*)d_in[0];
    // d_in[1] = Wq, d_in[2] = Wk : provably irrelevant (softmax rowsum == 1)
    const float* Wv  = (const float*)d_in[3];
    const float* Wo  = (const float*)d_in[4];
    const float* bo  = (const float*)d_in[5];
    float* out       = (float*)d_out;

    float* Wv_sum = (float*)d_ws;                       // 128*2048*4   = 1 MB
    float* vsum   = (float*)((char*)d_ws + (1 << 20));  // 32768*128*4  = 16 MB

    // A: fold Wv over heads
    fold_wv_kernel<<<(HD * EMB) / 256, 256, 0, stream>>>(Wv, Wv_sum);

    // B: vsum = x @ Wv_sum^T   (2048 waves, 8 waves/block)
    gemm1_wmma_kernel<<<(MROWS / 16) / 8, 256, 0, stream>>>(x, Wv_sum, vsum);

    // C: out = broadcast16( P @ Wo^T + bo ),  P = vsum as [2048][2048]
    gemm2_wmma_kernel<<<(128 * 32) / 8, 256, 0, stream>>>(vsum, Wo, bo, out);
}